// STPAttentionV4_82094004896538
// MI455X (gfx1250) — compile-verified
//
#include <hip/hip_runtime.h>
#include <hip/hip_bf16.h>

typedef __attribute__((ext_vector_type(2))) float v2f;
typedef __attribute__((ext_vector_type(8))) float v8f;

#define BM 128
#define BN 64
#define BK 16
#define BKP (BK + 4)   // pad: row stride 20 floats (80B) -> conflict-free, keeps 16B alignment

// ---------------------------------------------------------------------------
// CDNA5 async global->LDS copy (GLOBAL_LOAD_ASYNC_TO_LDS_B128, ASYNCcnt) with
// sync fallback.  Builtin signature (from hipcc diagnostic): the pointers are
// int-vector(4) typed, global side addrspace(1), LDS side addrspace(3).
// LDS generic pointers: low 32 bits are the LDS byte offset; global generic
// pointers are identity in addrspace(1).
// ---------------------------------------------------------------------------
#if __has_builtin(__builtin_amdgcn_global_load_async_to_lds_b128)
typedef int v4i_vs __attribute__((vector_size(16)));
typedef __attribute__((address_space(1))) v4i_vs as1_v4i;
typedef __attribute__((address_space(3))) v4i_vs as3_v4i;
__device__ __forceinline__ void async_copy_b128(void* lds, const void* gmem) {
    __builtin_amdgcn_global_load_async_to_lds_b128(
        (as1_v4i*)(unsigned long long)gmem,
        (as3_v4i*)(unsigned int)(unsigned long long)lds,
        0, 0);
}
__device__ __forceinline__ void async_wait0() {
#if __has_builtin(__builtin_amdgcn_s_wait_asynccnt)
    __builtin_amdgcn_s_wait_asynccnt(0);
#else
    asm volatile("s_wait_asynccnt 0x0" ::: "memory");
#endif
}
#else
__device__ __forceinline__ void async_copy_b128(void* lds, const void* gmem) {
    *reinterpret_cast<float4*>(lds) = *reinterpret_cast<const float4*>(gmem);
}
__device__ __forceinline__ void async_wait0() {}
#endif

// ---------------------------------------------------------------------------
// C[M,N] = A[M,K] * B[N,K]^T   (all fp32, M%128==0, N%64==0, K%16==0)
// fp32 WMMA: V_WMMA_F32_16X16X4_F32. Each wave -> 16 (M) x 64 (N) strip.
// Double-buffered LDS tiles staged with async global->LDS copies.
// ---------------------------------------------------------------------------
__global__ __launch_bounds__(256) void gemm_abt_f32(const float* __restrict__ A,
                                                    const float* __restrict__ B,
                                                    float* __restrict__ C,
                                                    int M, int N, int K) {
    __shared__ __align__(16) float As[2][BM][BKP];
    __shared__ __align__(16) float Bs[2][BN][BKP];

    const int tid   = threadIdx.x;
    const int wave  = tid >> 5;       // 0..7
    const int lane  = tid & 31;
    const int bm    = blockIdx.x * BM;
    const int bn    = blockIdx.y * BN;

    v8f acc[4];
#pragma unroll
    for (int j = 0; j < 4; ++j)
#pragma unroll
        for (int i = 0; i < 8; ++i) acc[j][i] = 0.0f;

    const int mrow = lane & 15;        // M index inside 16-row tile
    const int kh   = (lane >> 4) * 2;  // lanes 0-15 -> K {0,1}; 16-31 -> K {2,3}

    // A tile: 128x16 floats = 512 float4 (2/thread); B tile: 64x16 = 256 (1/thread)
    auto stage = [&](int bufi, int k0) {
#pragma unroll
        for (int i = 0; i < 2; ++i) {
            int idx = tid + i * 256;
            int r   = idx >> 2;
            int c4  = idx & 3;
            async_copy_b128(&As[bufi][r][c4 * 4],
                            &A[(size_t)(bm + r) * K + k0 + c4 * 4]);
        }
        {
            int r  = tid >> 2;
            int c4 = tid & 3;
            async_copy_b128(&Bs[bufi][r][c4 * 4],
                            &B[(size_t)(bn + r) * K + k0 + c4 * 4]);
        }
    };

    int buf = 0;
    stage(0, 0);

    for (int k0 = 0; k0 < K; k0 += BK) {
        async_wait0();          // this wave's staged writes for `buf` complete
        __syncthreads();        // all waves' writes visible / prev reads done
        if (k0 + BK < K) stage(buf ^ 1, k0 + BK);

#pragma unroll
        for (int kk = 0; kk < BK; kk += 4) {
            v2f a = *reinterpret_cast<const v2f*>(&As[buf][wave * 16 + mrow][kk + kh]);
#pragma unroll
            for (int j = 0; j < 4; ++j) {
                v2f b = *reinterpret_cast<const v2f*>(&Bs[buf][j * 16 + mrow][kk + kh]);
                acc[j] = __builtin_amdgcn_wmma_f32_16x16x4_f32(
                    false, a, false, b, (short)0, acc[j], false, false);
            }
        }
        buf ^= 1;
    }

    // C layout: VGPR i -> row (wave*16 + i + 8*(lane>=16)), col = bn + j*16 + (lane&15)
    const int rbase = bm + wave * 16 + ((lane >> 4) << 3);
#pragma unroll
    for (int j = 0; j < 4; ++j) {
        int col = bn + j * 16 + (lane & 15);
#pragma unroll
        for (int i = 0; i < 8; ++i)
            C[(size_t)(rbase + i) * N + col] = acc[j][i];
    }
}

// ---------------------------------------------------------------------------
// V_gs[row,h] = dot(x[row,:], W_vgs[h,:]) + base[h].  One wave per (row,h).
// ---------------------------------------------------------------------------
__global__ __launch_bounds__(256) void vgs_kernel(const float* __restrict__ x,
                                                  const float* __restrict__ Wv,
                                                  const float* __restrict__ base,
                                                  float* __restrict__ vgs, int rows) {
    int gwave = (int)((blockIdx.x * blockDim.x + threadIdx.x) >> 5);
    int lane  = threadIdx.x & 31;
    int h     = gwave & 15;
    int row   = gwave >> 4;
    if (row >= rows) return;
    const float* xr = x + (size_t)row * 1024;
    const float* wr = Wv + (size_t)h * 1024;
    float s = 0.f;
    for (int i = lane; i < 1024; i += 32) s += xr[i] * wr[i];
#pragma unroll
    for (int off = 16; off > 0; off >>= 1) s += __shfl_down(s, off, 32);
    if (lane == 0) vgs[(size_t)row * 16 + h] = s + base[h];
}

// ---------------------------------------------------------------------------
// In-place layernorm of k section of qkv: per (row,h) over 64 elems.
// qkv row layout: [q(16x64) | k(16x64) | v(16x64)] = 3072 floats.
// ---------------------------------------------------------------------------
__global__ __launch_bounds__(256) void ln_k_kernel(float* __restrict__ qkv, int rows) {
    int gwave = (int)((blockIdx.x * blockDim.x + threadIdx.x) >> 5);
    int lane  = threadIdx.x & 31;
    int h     = gwave & 15;
    int row   = gwave >> 4;
    if (row >= rows) return;
    float* kp = qkv + (size_t)row * 3072 + 1024 + h * 64;
    float a = kp[lane];
    float b = kp[lane + 32];
    float sum = a + b;
#pragma unroll
    for (int off = 16; off > 0; off >>= 1) sum += __shfl_down(sum, off, 32);
    sum = __shfl(sum, 0, 32);
    float mean = sum * (1.0f / 64.0f);
    float da = a - mean, db = b - mean;
    float vs = da * da + db * db;
#pragma unroll
    for (int off = 16; off > 0; off >>= 1) vs += __shfl_down(vs, off, 32);
    vs = __shfl(vs, 0, 32);
    float inv = rsqrtf(vs * (1.0f / 64.0f) + 1e-5f);
    kp[lane]      = da * inv;
    kp[lane + 32] = db * inv;
}

// ---------------------------------------------------------------------------
// Sequential scan.  Grid: 128 blocks = (b,h) x 4 d-groups; 8 independent
// waves/block, each wave owns 2 rows (d) of the 64x64 state F.
// Lane map: d_local = lane>>4 (0/1), e = (lane&15)*4  -> 4 state elems/lane.
// No barriers; y e-reduction via width-16 shuffles; t+1 loads pipelined.
// ---------------------------------------------------------------------------
__global__ __launch_bounds__(256) void scan_kernel(
    const float* __restrict__ qkv,   // (B*L, 3072)
    const float* __restrict__ vgs,   // (B*L, 16)
    const float* __restrict__ Wltm,  // (16, 64, 64)
    const float* __restrict__ V_T0, const float* __restrict__ beta_tau,
    const float* __restrict__ beta_gm, const float* __restrict__ C_ch,
    const float* __restrict__ gamma_, const float* __restrict__ alpha_ppd,
    const float* __restrict__ IC_thr,
    float* __restrict__ y,           // (B*L, 1024)
    int L) {
    const int bh     = blockIdx.x >> 2;   // 0..31
    const int dgroup = blockIdx.x & 3;
    const int b      = bh >> 4;
    const int h      = bh & 15;
    const int lane   = threadIdx.x & 31;
    const int wave   = threadIdx.x >> 5;  // 0..7
    const int d      = dgroup * 16 + wave * 2 + (lane >> 4);
    const int e      = (lane & 15) * 4;

    const float vt0   = V_T0[h];
    const float bt    = beta_tau[h];
    const float bg    = beta_gm[h];
    const float icc   = 1.0f / C_ch[h];
    const float gm    = gamma_[h];
    const float alpha = alpha_ppd[0];
    const float thr   = IC_thr[0];

    float w[4], F[4];
    const float* Wh = Wltm + ((size_t)h * 64 + d) * 64 + e;
#pragma unroll
    for (int i = 0; i < 4; ++i) { w[i] = Wh[i]; F[i] = 0.0f; }

    const size_t rowbase = (size_t)b * L;

    // prologue: load t = 0
    const float* qp0 = qkv + rowbase * 3072 + h * 64;
    float4 q4 = *reinterpret_cast<const float4*>(qp0 + e);
    float4 k4 = *reinterpret_cast<const float4*>(qp0 + 1024 + e);
    float  vd = qp0[2048 + d];
    float  s  = vgs[rowbase * 16 + h] - vt0;

    for (int t = 0; t < L; ++t) {
        // pipeline: issue loads for t+1 before computing t
        float4 qn = q4, kn = k4;
        float  vn = vd, sn = s;
        if (t + 1 < L) {
            const float* qp = qkv + (rowbase + t + 1) * 3072 + h * 64;
            qn = *reinterpret_cast<const float4*>(qp + e);
            kn = *reinterpret_cast<const float4*>(qp + 1024 + e);
            vn = qp[2048 + d];
            sn = vgs[(rowbase + t + 1) * 16 + h] - vt0;
        }

        const float qa[4] = {q4.x, q4.y, q4.z, q4.w};
        const float ka[4] = {k4.x, k4.y, k4.z, k4.w};
        float acc = 0.0f;
#pragma unroll
        for (int i = 0; i < 4; ++i) {
            float veff = s + w[i];
            float ex   = __expf(veff);
            bool  big  = veff > 20.0f;
            float sp   = big ? veff : __logf(1.0f + ex);      // softplus
            float sig  = big ? 1.0f : ex / (1.0f + ex);       // sigmoid
            float g    = bt * sp;
            float ret  = __expf(-g * icc);                    // retention
            float G    = bg * sp * sig;
            float a2   = alpha * (g - thr);
            float e2   = __expf(2.0f * a2);
            float sgn  = 1.0f - 2.0f / (e2 + 1.0f);           // tanh, overflow-safe
            F[i] = ret * F[i] + gm * sgn * G * (vd * ka[i]);
            acc += (w[i] + F[i]) * qa[i];
        }
        // reduce the 16 e-lane partials for this d (width-16 shuffle tree)
#pragma unroll
        for (int off = 8; off > 0; off >>= 1) acc += __shfl_down(acc, off, 16);
        if ((lane & 15) == 0)
            y[(rowbase + t) * 1024 + (size_t)h * 64 + d] = acc;

        q4 = qn; k4 = kn; vd = vn; s = sn;
    }
}

// ---------------------------------------------------------------------------
extern "C" void kernel_launch(void* const* d_in, const int* in_sizes, int n_in,
                              void* d_out, int out_size, void* d_ws, size_t ws_size,
                              hipStream_t stream) {
    const float* x        = (const float*)d_in[0];   // (2,2048,1024)
    const float* W_qkv    = (const float*)d_in[1];   // (3072,1024)
    const float* W_o      = (const float*)d_in[2];   // (1024,1024)
    const float* W_vgs    = (const float*)d_in[3];   // (16,1024)
    const float* W_LTM    = (const float*)d_in[4];   // (16,64,64)
    const float* V_gsb    = (const float*)d_in[5];   // (16)
    const float* V_T0     = (const float*)d_in[6];   // (16)
    const float* beta_tau = (const float*)d_in[7];   // (16)
    const float* beta_gm  = (const float*)d_in[8];   // (16)
    const float* C_ch     = (const float*)d_in[9];   // (16)
    const float* gamma_   = (const float*)d_in[10];  // (16)
    const float* alpha    = (const float*)d_in[11];  // (1)
    const float* icthr    = (const float*)d_in[12];  // (1)
    float* out = (float*)d_out;                      // (4096,1024)

    const int BL = 4096;        // B*L
    const int D  = 1024;
    const int L  = 2048;

    float* qkv = (float*)d_ws;                       // 4096*3072 f32 = 50.3 MB
    float* vgs = qkv + (size_t)BL * 3 * D;           // 4096*16
    float* yb  = vgs + (size_t)BL * 16;              // 4096*1024 = 16.8 MB

    // 1) qkv = x @ W_qkv^T   (fp32 WMMA, async-LDS double buffered)
    {
        dim3 grid(BL / BM, (3 * D) / BN);
        gemm_abt_f32<<<grid, 256, 0, stream>>>(x, W_qkv, qkv, BL, 3 * D, D);
    }
    // 2) V_gs = x @ W_vgs^T + base
    vgs_kernel<<<(BL * 16) / 8, 256, 0, stream>>>(x, W_vgs, V_gsb, vgs, BL);
    // 3) layernorm k in place
    ln_k_kernel<<<(BL * 16) / 8, 256, 0, stream>>>(qkv, BL);
    // 4) sequential plasticity scan -> y
    scan_kernel<<<128, 256, 0, stream>>>(qkv, vgs, W_LTM, V_T0, beta_tau,
                                         beta_gm, C_ch, gamma_, alpha, icthr,
                                         yb, L);
    // 5) out = y @ W_o^T   (fp32 WMMA)
    {
        dim3 grid(BL / BM, D / BN);
        gemm_abt_f32<<<grid, 256, 0, stream>>>(yb, W_o, out, BL, D, D);
    }
}